// SS2D_62723702390855
// MI455X (gfx1250) — compile-verified
//
#include <hip/hip_runtime.h>
#include <math.h>

#define D_MODEL  96
#define D_INNER  192
#define DT_RANK  6
#define D_STATE  16
#define K_DIRS   4
#define L_LEN    4096
#define C_ROWS   (DT_RANK + 2*D_STATE)   // 38 rows of x_dbl per direction
#define CHUNK    64
#define NCHUNK   (L_LEN / CHUNK)         // 64 chunks per scan

typedef float v2f __attribute__((ext_vector_type(2)));
typedef float v8f __attribute__((ext_vector_type(8)));

// Map scan-order index l (direction k) -> row-major spatial index.
__device__ __forceinline__ int posmap(int k, int l) {
    switch (k) {
        case 0:  return l;                               // row-major
        case 1:  return ((l & 63) << 6) | (l >> 6);      // transpose (w-major)
        case 2:  return (L_LEN - 1) - l;                 // reversed row-major
        default: { int m = (L_LEN - 1) - l;              // reversed transpose
                   return ((m & 63) << 6) | (m >> 6); }
    }
}

__device__ __forceinline__ v8f wmma_f32_k4(v2f a, v2f b, v8f c) {
    // V_WMMA_F32_16X16X4_F32: full-precision f32 matrix op on CDNA5
    return __builtin_amdgcn_wmma_f32_16x16x4_f32(false, a, false, b,
                                                 (short)0, c, false, false);
}

// ---------------- 1) xz = x @ in_proj_w^T : (4096x96)*(96x384) ----------------
__global__ void k_inproj(const float* __restrict__ x, const float* __restrict__ w,
                         float* __restrict__ xz) {
    int lane = threadIdx.x;
    int row  = lane & 15;        // M (for A) / N (for B) index within tile
    int kh   = lane >> 4;        // K half selector
    int m0 = blockIdx.x * 16;
    int n0 = blockIdx.y * 16;
    const float* ap = x + (size_t)(m0 + row) * D_MODEL + 2 * kh;
    const float* bp = w + (size_t)(n0 + row) * D_MODEL + 2 * kh;  // B[c][o] = w[o][c]
    v8f acc = {};
    #pragma unroll 4
    for (int k0 = 0; k0 < D_MODEL; k0 += 4) {
        v2f a, b;
        a.x = ap[k0]; a.y = ap[k0 + 1];
        b.x = bp[k0]; b.y = bp[k0 + 1];
        acc = wmma_f32_k4(a, b, acc);
    }
    int n = n0 + row;
    #pragma unroll
    for (int j = 0; j < 8; ++j) {
        int m = m0 + j + 8 * kh;
        xz[(size_t)m * (2 * D_INNER) + n] = acc[j];
    }
}

// ---------------- 2) depthwise 3x3 conv + bias + SiLU ----------------
__global__ void k_conv(const float* __restrict__ xz, const float* __restrict__ cw,
                       const float* __restrict__ cb, float* __restrict__ xc) {
    int t = blockIdx.x * blockDim.x + threadIdx.x;
    if (t >= L_LEN * D_INNER) return;
    int d = t % D_INNER;
    int l = t / D_INNER;
    int h = l >> 6, w = l & 63;
    float s = cb[d];
    #pragma unroll
    for (int dh = 0; dh < 3; ++dh) {
        int hh = h + dh - 1;
        if (hh < 0 || hh >= 64) continue;
        #pragma unroll
        for (int dw = 0; dw < 3; ++dw) {
            int ww = w + dw - 1;
            if (ww < 0 || ww >= 64) continue;
            s += xz[(size_t)((hh << 6) | ww) * (2 * D_INNER) + d] *
                 cw[(dh * 3 + dw) * D_INNER + d];
        }
    }
    xc[(size_t)l * D_INNER + d] = s / (1.0f + expf(-s));   // SiLU
}

// ---------------- 3) x_dbl[k] = x_proj_w[k] @ xs[k] : (38x192)*(192x4096) ----
__global__ void k_xdbl(const float* __restrict__ xpw, const float* __restrict__ xc,
                       float* __restrict__ xdbl) {
    int lane = threadIdx.x;
    int row  = lane & 15;
    int kh   = lane >> 4;
    int n0 = blockIdx.x * 16;      // l tile
    int m0 = blockIdx.y * 16;      // c tile (rows, padded 38 -> 48)
    int k  = blockIdx.z;           // direction
    int l  = n0 + row;
    int pos = posmap(k, l);
    const float* bp = xc + (size_t)pos * D_INNER + 2 * kh;   // B[d][l] = xc[pos(l)][d]
    int  arow = m0 + row;
    bool av   = arow < C_ROWS;
    const float* ap = xpw + (size_t)(k * C_ROWS + (av ? arow : 0)) * D_INNER + 2 * kh;
    v8f acc = {};
    for (int k0 = 0; k0 < D_INNER; k0 += 4) {
        v2f a, b;
        a.x = av ? ap[k0]     : 0.0f;
        a.y = av ? ap[k0 + 1] : 0.0f;
        b.x = bp[k0]; b.y = bp[k0 + 1];
        acc = wmma_f32_k4(a, b, acc);
    }
    #pragma unroll
    for (int j = 0; j < 8; ++j) {
        int c = m0 + j + 8 * kh;
        if (c < C_ROWS)
            xdbl[(size_t)(k * C_ROWS + c) * L_LEN + l] = acc[j];
    }
}

// ---------------- 4) delta = softplus(dt_w @ dts + dt_b) ----------------
__global__ void k_delta(const float* __restrict__ xdbl, const float* __restrict__ dtw,
                        const float* __restrict__ dtb, float* __restrict__ delta) {
    int t = blockIdx.x * blockDim.x + threadIdx.x;     // t = ((k*192)+d)*4096 + l
    if (t >= K_DIRS * D_INNER * L_LEN) return;
    int l = t & (L_LEN - 1);
    int d = (t >> 12) % D_INNER;
    int k = t / (L_LEN * D_INNER);
    float s = dtb[k * D_INNER + d];
    #pragma unroll
    for (int r = 0; r < DT_RANK; ++r)
        s += xdbl[(size_t)(k * C_ROWS + r) * L_LEN + l] *
             dtw[(k * D_INNER + d) * DT_RANK + r];
    delta[t] = (s > 20.0f) ? s : log1pf(expf(s));       // softplus
}

// ---------------- 5a) chunked scan pass A: per-chunk dA-product & partial state
// Thread index t = ((sid*NCHUNK)+chunk)*16 + n ; sid = k*192+d.
__global__ void k_scan_part(const float* __restrict__ xc, const float* __restrict__ xdbl,
                            const float* __restrict__ delta, const float* __restrict__ alg,
                            float* __restrict__ Pa, float* __restrict__ Ha) {
    int t   = blockIdx.x * blockDim.x + threadIdx.x;   // 786432 threads
    int n   = t & 15;
    int c   = (t >> 4) & (NCHUNK - 1);
    int sid = t >> 10;
    int d   = sid % D_INNER;
    int k   = sid / D_INNER;
    float A = -expf(alg[sid * D_STATE + n]);
    const float* Brow = xdbl + (size_t)(k * C_ROWS + DT_RANK + n) * L_LEN;
    const float* drow = delta + (size_t)sid * L_LEN;
    float h = 0.0f, P = 1.0f;
    int l0 = c * CHUNK;
    for (int i = 0; i < CHUNK; ++i) {
        int l    = l0 + i;
        int pos  = posmap(k, l);
        float xv = xc[(size_t)pos * D_INNER + d];
        float dl = drow[l];
        float dA = expf(dl * A);
        h = dA * h + dl * Brow[l] * xv;
        P *= dA;
    }
    Pa[t] = P;
    Ha[t] = h;
}

// ---------------- 5b) pass B: serial carry over chunk summaries (64 steps)
__global__ void k_scan_carry(const float* __restrict__ Pa, const float* __restrict__ Ha,
                             float* __restrict__ Hin) {
    int t   = blockIdx.x * blockDim.x + threadIdx.x;   // 12288 threads = (sid, n)
    int n   = t & 15;
    int sid = t >> 4;
    float h = 0.0f;
    for (int c = 0; c < NCHUNK; ++c) {
        int idx = ((sid * NCHUNK) + c) * 16 + n;
        Hin[idx] = h;
        h = Pa[idx] * h + Ha[idx];
    }
}

// ---------------- 5c) pass C: re-run chunk with carry-in, emit y per position
__global__ void k_scan_apply(const float* __restrict__ xc, const float* __restrict__ xdbl,
                             const float* __restrict__ delta, const float* __restrict__ alg,
                             const float* __restrict__ Ds, const float* __restrict__ Hin,
                             float* __restrict__ yp) {
    int t   = blockIdx.x * blockDim.x + threadIdx.x;   // 786432 threads
    int n   = t & 15;
    int c   = (t >> 4) & (NCHUNK - 1);
    int sid = t >> 10;
    int d   = sid % D_INNER;
    int k   = sid / D_INNER;
    float A  = -expf(alg[sid * D_STATE + n]);
    float Dv = Ds[sid];
    const float* Brow = xdbl + (size_t)(k * C_ROWS + DT_RANK + n) * L_LEN;
    const float* Crow = xdbl + (size_t)(k * C_ROWS + DT_RANK + D_STATE + n) * L_LEN;
    const float* drow = delta + (size_t)sid * L_LEN;
    float* yrow = yp + (size_t)sid * L_LEN;
    float h = Hin[t];
    int l0 = c * CHUNK;
    for (int i = 0; i < CHUNK; ++i) {
        int l    = l0 + i;
        int pos  = posmap(k, l);
        float xv = xc[(size_t)pos * D_INNER + d];
        float dl = drow[l];
        h = expf(dl * A) * h + dl * Brow[l] * xv;
        float v = h * Crow[l];
        v += __shfl_xor(v, 1, 16);
        v += __shfl_xor(v, 2, 16);
        v += __shfl_xor(v, 4, 16);
        v += __shfl_xor(v, 8, 16);
        if (n == 0) yrow[l] = v + Dv * xv;
    }
}

// ---------------- 6) sum over directions + LayerNorm + SiLU gate ----------------
__global__ void k_combine(const float* __restrict__ yp, const float* __restrict__ xz,
                          const float* __restrict__ lng, const float* __restrict__ lnb,
                          float* __restrict__ yg) {
    __shared__ float s1[D_INNER];
    __shared__ float s2[D_INNER];
    int g = blockIdx.x;            // row-major spatial index
    int d = threadIdx.x;           // 0..191
    int l1 = ((g & 63) << 6) | (g >> 6);
    int l2 = (L_LEN - 1) - g;
    int l3 = (L_LEN - 1) - l1;
    float y = yp[(size_t)(0 * D_INNER + d) * L_LEN + g]
            + yp[(size_t)(1 * D_INNER + d) * L_LEN + l1]
            + yp[(size_t)(2 * D_INNER + d) * L_LEN + l2]
            + yp[(size_t)(3 * D_INNER + d) * L_LEN + l3];
    s1[d] = y; s2[d] = y * y;
    __syncthreads();
    for (int s = 96; s >= 3; s >>= 1) {
        if (d < s) { s1[d] += s1[d + s]; s2[d] += s2[d + s]; }
        __syncthreads();
    }
    float sum1 = s1[0] + s1[1] + s1[2];
    float sum2 = s2[0] + s2[1] + s2[2];
    float mu  = sum1 * (1.0f / D_INNER);
    float var = sum2 * (1.0f / D_INNER) - mu * mu;
    float yn  = (y - mu) * rsqrtf(var + 1e-5f) * lng[d] + lnb[d];
    float z   = xz[(size_t)g * (2 * D_INNER) + D_INNER + d];
    yg[(size_t)g * D_INNER + d] = yn * (z / (1.0f + expf(-z)));
}

// ---------------- 7) out = yg @ out_w^T : (4096x192)*(192x96) ----------------
__global__ void k_out(const float* __restrict__ yg, const float* __restrict__ ow,
                      float* __restrict__ out) {
    int lane = threadIdx.x;
    int row  = lane & 15;
    int kh   = lane >> 4;
    int m0 = blockIdx.x * 16;
    int n0 = blockIdx.y * 16;
    const float* ap = yg + (size_t)(m0 + row) * D_INNER + 2 * kh;
    const float* bp = ow + (size_t)(n0 + row) * D_INNER + 2 * kh;  // B[d][o] = ow[o][d]
    v8f acc = {};
    for (int k0 = 0; k0 < D_INNER; k0 += 4) {
        v2f a, b;
        a.x = ap[k0]; a.y = ap[k0 + 1];
        b.x = bp[k0]; b.y = bp[k0 + 1];
        acc = wmma_f32_k4(a, b, acc);
    }
    int n = n0 + row;
    #pragma unroll
    for (int j = 0; j < 8; ++j)
        out[(size_t)(m0 + j + 8 * kh) * D_MODEL + n] = acc[j];
}

extern "C" void kernel_launch(void* const* d_in, const int* in_sizes, int n_in,
                              void* d_out, int out_size, void* d_ws, size_t ws_size,
                              hipStream_t stream) {
    const float* x   = (const float*)d_in[0];
    const float* ipw = (const float*)d_in[1];
    const float* cw  = (const float*)d_in[2];
    const float* cb  = (const float*)d_in[3];
    const float* xpw = (const float*)d_in[4];
    const float* dtw = (const float*)d_in[5];
    const float* dtb = (const float*)d_in[6];
    const float* alg = (const float*)d_in[7];
    const float* ds  = (const float*)d_in[8];
    const float* lng = (const float*)d_in[9];
    const float* lnb = (const float*)d_in[10];
    const float* ow  = (const float*)d_in[11];
    float* out = (float*)d_out;

    char* ws = (char*)d_ws;
    size_t off = 0;
    const size_t NSCAN = (size_t)K_DIRS * D_INNER * D_STATE * NCHUNK;  // 786432
    float* xz    = (float*)(ws + off); off += (size_t)L_LEN * 2 * D_INNER * 4;
    float* xc    = (float*)(ws + off); off += (size_t)L_LEN * D_INNER * 4;
    float* xdbl  = (float*)(ws + off); off += (size_t)K_DIRS * C_ROWS * L_LEN * 4;
    float* delta = (float*)(ws + off); off += (size_t)K_DIRS * D_INNER * L_LEN * 4;
    float* yp    = (float*)(ws + off); off += (size_t)K_DIRS * D_INNER * L_LEN * 4;
    float* yg    = (float*)(ws + off); off += (size_t)L_LEN * D_INNER * 4;
    float* Pa    = (float*)(ws + off); off += NSCAN * 4;
    float* Ha    = (float*)(ws + off); off += NSCAN * 4;
    float* Hin   = (float*)(ws + off); off += NSCAN * 4;
    (void)ws_size; (void)in_sizes; (void)n_in; (void)out_size;

    k_inproj    <<<dim3(L_LEN / 16, (2 * D_INNER) / 16), 32, 0, stream>>>(x, ipw, xz);
    k_conv      <<<(L_LEN * D_INNER + 255) / 256, 256, 0, stream>>>(xz, cw, cb, xc);
    k_xdbl      <<<dim3(L_LEN / 16, 3, K_DIRS), 32, 0, stream>>>(xpw, xc, xdbl);
    k_delta     <<<(K_DIRS * D_INNER * L_LEN) / 256, 256, 0, stream>>>(xdbl, dtw, dtb, delta);
    k_scan_part <<<(unsigned)(NSCAN / 256), 256, 0, stream>>>(xc, xdbl, delta, alg, Pa, Ha);
    k_scan_carry<<<(K_DIRS * D_INNER * D_STATE) / 256, 256, 0, stream>>>(Pa, Ha, Hin);
    k_scan_apply<<<(unsigned)(NSCAN / 256), 256, 0, stream>>>(xc, xdbl, delta, alg, ds, Hin, yp);
    k_combine   <<<L_LEN, D_INNER, 0, stream>>>(yp, xz, lng, lnb, yg);
    k_out       <<<dim3(L_LEN / 16, D_MODEL / 16), 32, 0, stream>>>(yg, ow, out);
}